// LocalityConvertor_61168924230005
// MI455X (gfx1250) — compile-verified
//
#include <hip/hip_runtime.h>
#include <stdint.h>

#define B_  16
#define C_  64
#define H_  256
#define W_  256
#define HW_ (H_ * W_)

// ---------- float -> monotonic sortable key and back ----------
__device__ __forceinline__ unsigned f2key(float f) {
    unsigned u = __float_as_uint(f);
    return (u & 0x80000000u) ? ~u : (u | 0x80000000u);
}
__device__ __forceinline__ float key2f(unsigned key) {
    unsigned u = (key & 0x80000000u) ? (key ^ 0x80000000u) : ~key;
    return __uint_as_float(u);
}

__device__ __forceinline__ void async_copy16(unsigned ldsOff, const void* gaddr) {
    // gfx1250: 16B per lane, global -> LDS, tracked by ASYNCcnt
    asm volatile("global_load_async_to_lds_b128 %0, %1, off"
                 :: "v"(ldsOff), "v"((unsigned long long)(uintptr_t)gaddr)
                 : "memory");
}

// ============================================================
// Kernel 1: channel max/argmax, double-buffered async global->LDS.
// Block = one (b, h) row; 256 threads; 16 groups x 4 channels.
// Pipeline: issue group g+1 into the other LDS buffer, then
// s_wait_asynccnt 1 (only the older DMA must retire) -> barrier ->
// fold the 4 staged channel rows. One DMA always in flight per wave.
// ============================================================
__global__ __launch_bounds__(256) void k_max_pack(const float* __restrict__ x,
                                                  unsigned long long* __restrict__ recs)
{
    __shared__ __align__(16) float tile[2][4][W_];   // 8 KB ping-pong
    const int t  = threadIdx.x;
    const int bh = blockIdx.x;          // 0 .. B_*H_-1
    const int b  = bh >> 8;             // H_ == 256
    const int h  = bh & (H_ - 1);

    const float* xb = x + (size_t)b * C_ * HW_ + (size_t)h * W_;
    const int cSub = t >> 6;            // 0..3   (channel within group)
    const int wq   = (t & 63) << 2;     // w base of my 16B quad
    const unsigned lds0 = (unsigned)(uintptr_t)&tile[0][cSub][wq];
    const unsigned lds1 = (unsigned)(uintptr_t)&tile[1][cSub][wq];
    const float* gsrc = xb + (size_t)cSub * HW_ + wq;   // group stride: 4*HW_ floats

    float best  = -3.402823466e38f;
    int   bestc = 0;

    // prologue: group 0 -> buffer 0
    async_copy16(lds0, gsrc);

    for (int g = 0; g < 16; ++g) {
        if (g < 15) {
            // issue group g+1 into the other buffer, then wait only for
            // the older transfer (group g) to retire
            async_copy16((g & 1) ? lds0 : lds1, gsrc + (size_t)(g + 1) * 4 * HW_);
            asm volatile("s_wait_asynccnt 1" ::: "memory");
        } else {
            asm volatile("s_wait_asynccnt 0" ::: "memory");
        }
        __syncthreads();                 // group g fully staged for all waves
        const int buf = g & 1;
#pragma unroll
        for (int j = 0; j < 4; ++j) {
            float v = tile[buf][j][t];
            if (v > best) { best = v; bestc = g * 4 + j; }  // strict > == first-occurrence argmax
        }
        __syncthreads();                 // all waves done reading buf before it is reused
    }

    const unsigned p = (unsigned)(h * W_ + t);   // spatial flat index (16 bits)
    unsigned long long rec = ((unsigned long long)f2key(best) << 32)
                           | ((unsigned long long)p << 8)
                           | (unsigned)bestc;
    recs[(size_t)b * HW_ + p] = rec;             // in p-order -> stable sort baseline
}

// ============================================================
// Kernel 2: per-batch stable LSD radix-16 sort (8 passes over key
// bits [32..63]) + decode/emit. One block per batch, 512 threads,
// 128 elements per thread. LDS: 16x512 count matrix (digit-major)
// + 512 chunk sums; flat exclusive scan gives stable scatter bases.
// Per-lane chunks are streamed sequentially (32 live cachelines per
// wave -> cache-resident), so reads run at full bandwidth.
// ============================================================
#define NT  512
#define EPT 128   // NT * EPT == HW_

__global__ __launch_bounds__(NT) void k_sort_emit(unsigned long long* __restrict__ A,
                                                  unsigned long long* __restrict__ Bb,
                                                  float4* __restrict__ out)
{
    __shared__ unsigned cnt[16 * NT];   // 32 KB, flat index = digit*NT + thread
    __shared__ unsigned csum[NT];       // 2 KB
    const int t      = threadIdx.x;
    const int batch  = blockIdx.x;
    const size_t base  = (size_t)batch * HW_;
    const size_t cbase = base + (size_t)t * EPT;

    for (int pass = 0; pass < 8; ++pass) {
        const unsigned long long* src = (pass & 1) ? Bb : A;
        unsigned long long*       dst = (pass & 1) ? A  : Bb;
        const int shift = 32 + pass * 4;

        // 1) per-thread digit histogram of my contiguous chunk
        unsigned lc[16];
#pragma unroll
        for (int d = 0; d < 16; ++d) lc[d] = 0;
        __builtin_prefetch(&src[cbase], 0, 3);       // global_prefetch_b8
        for (int i = 0; i < EPT; ++i) {
            unsigned d = (unsigned)(src[cbase + i] >> shift) & 15u;
            lc[d]++;
        }
#pragma unroll
        for (int d = 0; d < 16; ++d) cnt[d * NT + t] = lc[d];
        __syncthreads();

        // 2) flat exclusive scan of all 8192 counters (digit-major order)
        unsigned run = 0;
        unsigned vals[16];
#pragma unroll
        for (int j = 0; j < 16; ++j) {
            unsigned v = cnt[t * 16 + j];
            vals[j] = run;
            run += v;
        }
        csum[t] = run;
        __syncthreads();
        for (int off = 1; off < NT; off <<= 1) {     // Hillis-Steele over chunk sums
            unsigned v = (t >= off) ? csum[t - off] : 0u;
            __syncthreads();
            csum[t] += v;
            __syncthreads();
        }
        const unsigned exBase = csum[t] - run;       // exclusive chunk base
#pragma unroll
        for (int j = 0; j < 16; ++j) cnt[t * 16 + j] = vals[j] + exBase;
        __syncthreads();

        // 3) stable ordered scatter
        unsigned off16[16];
#pragma unroll
        for (int d = 0; d < 16; ++d) off16[d] = cnt[d * NT + t];
        for (int i = 0; i < EPT; ++i) {
            unsigned long long rec = src[cbase + i];
            unsigned d = (unsigned)(rec >> shift) & 15u;
            dst[base + off16[d]] = rec;
            off16[d]++;
        }
        __syncthreads();
    }

    // 8 (even) passes -> sorted records are back in A. Decode + emit.
    for (int i = 0; i < EPT; ++i) {
        const size_t j = (size_t)t * EPT + i;
        const unsigned long long rec = A[base + j];
        const unsigned key = (unsigned)(rec >> 32);
        const unsigned p   = (unsigned)(rec >> 8) & 0xFFFFu;
        const unsigned c   = (unsigned)rec & 0xFFu;
        float4 o;
        o.x = key2f(key);                          // value
        o.y = (float)(c * (unsigned)HW_ + p);      // pool flat index (< 2^24, exact)
        o.z = (float)(p >> 8);                     // row
        o.w = (float)(p & (W_ - 1));               // col
        out[base + j] = o;
    }
}

extern "C" void kernel_launch(void* const* d_in, const int* in_sizes, int n_in,
                              void* d_out, int out_size, void* d_ws, size_t ws_size,
                              hipStream_t stream) {
    (void)in_sizes; (void)n_in; (void)out_size; (void)ws_size;
    const float* x = (const float*)d_in[0];
    unsigned long long* A  = (unsigned long long*)d_ws;                 // 8 MB
    unsigned long long* Bb = A + (size_t)B_ * HW_;                      // 8 MB
    k_max_pack<<<B_ * H_, 256, 0, stream>>>(x, A);
    k_sort_emit<<<B_, NT, 0, stream>>>(A, Bb, (float4*)d_out);
}